// DualFlowTransformer_74277164417545
// MI455X (gfx1250) — compile-verified
//
#include <hip/hip_runtime.h>
#include <math.h>

// ---------------------------------------------------------------------------
// CDNA5 (gfx1250) WMMA helpers: f32 16x16x4, wave32.
// A: 16x4: per lane (m = lane&15, khi = lane>>4): a.x = A[m][k0+2*khi], a.y = A[m][k0+2*khi+1]
// B: 4x16: b.x = B[k0+2*khi][n], b.y = B[k0+2*khi+1][n], n = lane&15
// C/D: 8 VGPRs: c[r] = C[r + 8*(lane>>4)][lane&15]
// ---------------------------------------------------------------------------
typedef float v2f __attribute__((ext_vector_type(2)));
typedef float v8f __attribute__((ext_vector_type(8)));

__device__ __forceinline__ v8f wmma4(v2f a, v2f b, v8f c) {
    return __builtin_amdgcn_wmma_f32_16x16x4_f32(false, a, false, b, (short)0, c, false, false);
}

#define NN 307
#define TT 256
#define HH 64
#define AP_R 320          // padded Anorm rows (20 M-tiles x 16)
#define AP_C 308          // padded Anorm cols (K padded to mult of 4)

// ===========================================================================
// Zero-padded Anorm image so the scan's WMMA A-fragments are unconditional
// aligned 8-byte loads (no EXEC-masked scalar loads in the hot loop).
// ===========================================================================
__global__ void pad_anorm_kernel(const float* __restrict__ A, float* __restrict__ Apad)
{
    int idx = blockIdx.x * 256 + threadIdx.x;
    if (idx >= AP_R * AP_C) return;
    int r = idx / AP_C, c = idx - r * AP_C;
    Apad[idx] = (r < NN && c < NN) ? A[r * NN + c] : 0.f;
}

// ===========================================================================
// Phase 1: persistent recurrent scan. One block per direction (grid.x = 2),
// 1024 threads, h state lives in LDS for all 256 steps. Anorm@msg_in is done
// with WMMA f32 16x16x4 (20 M-tiles x 5 N-tiles, K=308 padded), with the
// fragment loads software-pipelined one K-step ahead.
// LDS layout (floats): smi[308*80] | sam[320*80] | sh[307*64] | smix[307*4]
// (sam padded to 320 rows so the result store-back needs no row guard)
// ===========================================================================
#define CELL_LDS_FLOATS (308*80 + 320*80 + NN*HH + NN*4)

__global__ __launch_bounds__(1024, 1)
void cell_scan_kernel(const float* __restrict__ x_seq, const float* __restrict__ m_seq,
                      const float* __restrict__ tf_seq, const float* __restrict__ tj_seq,
                      const float* __restrict__ Apad,
                      const float* __restrict__ W0_, const float* __restrict__ b0_,
                      const float* __restrict__ W1_,
                      const float* __restrict__ mixW_, const float* __restrict__ mixb_,
                      const float* __restrict__ Wih_, const float* __restrict__ Whh_,
                      const float* __restrict__ bih_, const float* __restrict__ bhh_,
                      const float* __restrict__ outW_, const float* __restrict__ outb_,
                      const float* __restrict__ jamW_, const float* __restrict__ jamb_,
                      float* __restrict__ hsF, float* __restrict__ hsB,
                      float* __restrict__ pfb, float* __restrict__ pbb,
                      float* __restrict__ jfb, float* __restrict__ jbb)
{
    const int d   = blockIdx.x;          // 0 = forward, 1 = backward
    const int tid = threadIdx.x;

    const float* W0   = W0_   + d * 4 * HH * 67;
    const float* b0   = b0_   + d * 4 * HH;
    const float* W1   = W1_   + d * 4 * HH * 67;
    const float* mixW = mixW_ + d * 4 * HH;
    const float* mixb = mixb_ + d * 4;
    const float* Wih  = Wih_  + d * 192 * 66;
    const float* Whh  = Whh_  + d * 192 * 64;
    const float* bih  = bih_  + d * 192;
    const float* bhh  = bhh_  + d * 192;
    const float* outW = outW_ + d * HH;
    const float* outb = outb_ + d;
    const float* jamW = jamW_ + d * HH;
    const float* jamb = jamb_ + d;

    float* hs   = d ? hsB : hsF;
    float* pbuf = d ? pbb : pfb;
    float* jbuf = d ? jbb : jfb;

    extern __shared__ float lds[];
    float* smi  = lds;                 // msg_in / inp : 308 rows x 80 pitch
    float* sam  = smi + 308 * 80;      // am           : 320 rows x 80 pitch (padded)
    float* sh   = sam + 320 * 80;      // h            : 307 x 64
    float* smix = sh  + NN * HH;       // mix          : 307 x 4

    for (int i = tid; i < NN * HH; i += 1024) sh[i] = 0.f;
    __syncthreads();

    const int lane = tid & 31, wid = tid >> 5;
    const int mrow = lane & 15, khi = lane >> 4;

    for (int t = 0; t < TT; ++t) {
        const int tt = d ? (TT - 1 - t) : t;

        // ---- build msg_in = [h | m | tf | tj], zero pad row 307 -----------
        for (int e = tid; e < NN * 67; e += 1024) {
            int j = e / 67, i = e - j * 67;
            float v;
            if      (i < 64)  v = sh[j * HH + i];
            else if (i == 64) v = m_seq[j * TT + tt];
            else if (i == 65) v = tf_seq[j * TT + tt];
            else              v = tj_seq[j * TT + tt];
            smi[j * 80 + i] = v;
        }
        if (tid < 80) smi[NN * 80 + tid] = 0.f;
        __syncthreads();

        // ---- am = Anorm @ msg_in  via WMMA, loads pipelined one step -----
        for (int tile = wid; tile < 100; tile += 32) {
            int tm = tile / 5, tn = tile - tm * 5;
            int n0 = tm * 16, c0 = tn * 16;
            const float* arow = Apad + (size_t)(n0 + mrow) * AP_C;
            const int bcol = c0 + mrow;
            v8f acc = {};
            int ka = 2 * khi;
            v2f a = *(const v2f*)(arow + ka);
            v2f b;
            b.x = smi[ka * 80 + bcol];
            b.y = smi[(ka + 1) * 80 + bcol];
            for (int k0 = 0; k0 < 304; k0 += 4) {
                int kn = k0 + 4 + 2 * khi;
                v2f a2 = *(const v2f*)(arow + kn);
                v2f b2;
                b2.x = smi[kn * 80 + bcol];
                b2.y = smi[(kn + 1) * 80 + bcol];
                acc = wmma4(a, b, acc);
                a = a2; b = b2;
            }
            acc = wmma4(a, b, acc);
            for (int r = 0; r < 8; ++r)
                sam[(n0 + r + 8 * khi) * 80 + bcol] = acc[r];
        }
        // ---- mix logits (uses only h, independent of am) -----------------
        for (int e = tid; e < NN * 4; e += 1024) {
            int n = e >> 2, p = e & 3;
            float s = mixb[p];
            const float* hr = sh + n * HH;
            const float* wr = mixW + p * HH;
            for (int c = 0; c < HH; ++c) s += hr[c] * wr[c];
            smix[e] = s;
        }
        __syncthreads();

        // ---- softmax over p=4 --------------------------------------------
        if (tid < NN) {
            float* mr = smix + tid * 4;
            float mx = fmaxf(fmaxf(mr[0], mr[1]), fmaxf(mr[2], mr[3]));
            float e0 = __expf(mr[0] - mx), e1 = __expf(mr[1] - mx);
            float e2 = __expf(mr[2] - mx), e3 = __expf(mr[3] - mx);
            float inv = 1.f / (e0 + e1 + e2 + e3);
            mr[0] = e0 * inv; mr[1] = e1 * inv; mr[2] = e2 * inv; mr[3] = e3 * inv;
        }
        __syncthreads();

        // ---- msg[n][o] = sum_p mix * tanh(msgin.W0 + b0 + am.W1) ---------
        float regv[20];
        for (int k = 0; k < 20; ++k) {
            int e = tid + k * 1024;
            if (e < NN * HH) {
                int n = e >> 6, o = e & 63;
                const float* mir = smi + n * 80;
                const float* amr = sam + n * 80;
                float val = 0.f;
                for (int p = 0; p < 4; ++p) {
                    float s = b0[p * HH + o];
                    const float* w0r = W0 + (p * HH + o) * 67;
                    const float* w1r = W1 + (p * HH + o) * 67;
                    for (int i = 0; i < 67; ++i) s += mir[i] * w0r[i] + amr[i] * w1r[i];
                    val += smix[n * 4 + p] * tanhf(s);
                }
                regv[k] = val;
            }
        }
        __syncthreads();
        for (int k = 0; k < 20; ++k) {
            int e = tid + k * 1024;
            if (e < NN * HH) { int n = e >> 6, o = e & 63; smi[n * 80 + o] = regv[k]; }
        }
        if (tid < NN) {
            float mt = m_seq[tid * TT + tt];
            smi[tid * 80 + 64] = x_seq[tid * TT + tt] * mt;
            smi[tid * 80 + 65] = mt;
        }
        __syncthreads();

        // ---- GRU gates + state update ------------------------------------
        for (int k = 0; k < 20; ++k) {
            int e = tid + k * 1024;
            if (e < NN * HH) {
                int n = e >> 6, c = e & 63;
                const float* inr = smi + n * 80;
                const float* hr  = sh + n * HH;
                float ir = bih[c], iz = bih[64 + c], in_ = bih[128 + c];
                const float* wr = Wih + c * 66;
                const float* wz = Wih + (64 + c) * 66;
                const float* wn = Wih + (128 + c) * 66;
                for (int i = 0; i < 66; ++i) {
                    float xi = inr[i];
                    ir += xi * wr[i]; iz += xi * wz[i]; in_ += xi * wn[i];
                }
                float hrv = bhh[c], hz = bhh[64 + c], hn = bhh[128 + c];
                const float* ur = Whh + c * 64;
                const float* uz = Whh + (64 + c) * 64;
                const float* un = Whh + (128 + c) * 64;
                for (int j = 0; j < HH; ++j) {
                    float hj = hr[j];
                    hrv += hj * ur[j]; hz += hj * uz[j]; hn += hj * un[j];
                }
                float r  = 1.f / (1.f + __expf(-(ir + hrv)));
                float z  = 1.f / (1.f + __expf(-(iz + hz)));
                float nn = tanhf(in_ + r * hn);
                float hold = hr[c];
                float hnew = (1.f - z) * nn + z * hold;
                regv[k] = hnew + 0.1f * hold;     // h2
            }
        }
        __syncthreads();
        for (int k = 0; k < 20; ++k) {
            int e = tid + k * 1024;
            if (e < NN * HH) {
                int n = e >> 6, c = e & 63;
                sh[n * HH + c] = regv[k];
                hs[((size_t)n * TT + tt) * HH + c] = regv[k];
            }
        }
        __syncthreads();

        // ---- per-node pred / jam heads -----------------------------------
        if (tid < NN) {
            const float* hr = sh + tid * HH;
            float p = outb[0], j = jamb[0];
            for (int c = 0; c < HH; ++c) { p += hr[c] * outW[c]; j += hr[c] * jamW[c]; }
            pbuf[tid * TT + tt] = p;
            jbuf[tid * TT + tt] = j;
        }
        __syncthreads();
    }
}

// ===========================================================================
// Generic Y = X @ W^T (+bias)(+residual)(+relu) via WMMA. X:(M,K) W:(Nc,K).
// Each wave computes a 16(M) x 64(N) strip: one A fragment feeds 4 WMMAs per
// K-step (4 independent accumulators), and the next K-step's 5 fragment loads
// are issued before the current step's WMMAs. Requires M%16, K%4, Nc%64 == 0.
// ===========================================================================
__global__ __launch_bounds__(256)
void gemm_xwt_kernel(const float* __restrict__ X, const float* __restrict__ W,
                     const float* __restrict__ bias, const float* __restrict__ resid,
                     float* __restrict__ Y, int M, int K, int Nc, int relu)
{
    const int lane = threadIdx.x & 31, wid = threadIdx.x >> 5;
    const int mrow = lane & 15, khi = lane >> 4;
    const int tilesNq = Nc >> 6;                 // groups of 4 n-tiles
    const int tiles   = (M >> 4) * tilesNq;
    int tile = blockIdx.x * 8 + wid;
    if (tile >= tiles) return;
    int tm = tile / tilesNq, tq = tile - tm * tilesNq;

    const float* arow  = X + (size_t)(tm * 16 + mrow) * K;
    const float* brow0 = W + (size_t)(tq * 64 + mrow) * K;
    const size_t bs = (size_t)16 * K;
    v8f acc0 = {}, acc1 = {}, acc2 = {}, acc3 = {};

    int ka = 2 * khi;
    v2f a  = *(const v2f*)(arow + ka);
    v2f b0 = *(const v2f*)(brow0 + ka);
    v2f b1 = *(const v2f*)(brow0 + bs + ka);
    v2f b2 = *(const v2f*)(brow0 + 2 * bs + ka);
    v2f b3 = *(const v2f*)(brow0 + 3 * bs + ka);
    for (int k0 = 0; k0 < K - 4; k0 += 4) {
        int kn = k0 + 4 + 2 * khi;
        v2f an  = *(const v2f*)(arow + kn);
        v2f c0v = *(const v2f*)(brow0 + kn);
        v2f c1v = *(const v2f*)(brow0 + bs + kn);
        v2f c2v = *(const v2f*)(brow0 + 2 * bs + kn);
        v2f c3v = *(const v2f*)(brow0 + 3 * bs + kn);
        acc0 = wmma4(a, b0, acc0);
        acc1 = wmma4(a, b1, acc1);
        acc2 = wmma4(a, b2, acc2);
        acc3 = wmma4(a, b3, acc3);
        a = an; b0 = c0v; b1 = c1v; b2 = c2v; b3 = c3v;
    }
    acc0 = wmma4(a, b0, acc0);
    acc1 = wmma4(a, b1, acc1);
    acc2 = wmma4(a, b2, acc2);
    acc3 = wmma4(a, b3, acc3);

    v8f accs[4] = {acc0, acc1, acc2, acc3};
    for (int q = 0; q < 4; ++q) {
        int col = tq * 64 + q * 16 + mrow;
        float bv = bias ? bias[col] : 0.f;
        for (int r = 0; r < 8; ++r) {
            int row = tm * 16 + r + 8 * khi;
            size_t off = (size_t)row * Nc + col;
            float v = accs[q][r] + bv;
            if (resid) v += resid[off];
            if (relu)  v = fmaxf(v, 0.f);
            Y[off] = v;
        }
    }
}

// ===========================================================================
// Fused attention: one block per (node, head). Scores (256x256, pitch 258),
// Q/K/V (256x16) staged in LDS (~313 KB -> needs CDNA5's 320 KB WGP LDS).
// score-GEMM -> row softmax -> AV-GEMM, all WMMA; AV loads pipelined.
// ===========================================================================
#define ATTN_LDS_FLOATS (256*258 + 3*256*16)

__global__ __launch_bounds__(256)
void attn_kernel(const float* __restrict__ qkv, float* __restrict__ obuf)
{
    const int nh = blockIdx.x;            // n*4 + head
    const int n = nh >> 2, h = nh & 3;
    const int tid = threadIdx.x;

    extern __shared__ float lds[];
    float* S  = lds;                      // 256 x 258
    float* qs = S + 256 * 258;            // 256 x 16
    float* ks = qs + 4096;
    float* vs = ks + 4096;

    const size_t base = (size_t)n * TT;
    for (int e = tid; e < 4096; e += 256) {
        int t = e >> 4, dd = e & 15;
        size_t o = (base + t) * 192 + h * 16 + dd;
        qs[e] = qkv[o];
        ks[e] = qkv[o + 64];
        vs[e] = qkv[o + 128];
    }
    __syncthreads();

    const int lane = tid & 31, wid = tid >> 5;
    const int mrow = lane & 15, khi = lane >> 4;

    // S = Q @ K^T * (1/sqrt(16))
    for (int tile = wid; tile < 256; tile += 8) {
        int tm = tile >> 4, tn = tile & 15;
        v8f acc = {};
        for (int k0 = 0; k0 < 16; k0 += 4) {
            int ka = k0 + 2 * khi;
            v2f a = *(const v2f*)(qs + (tm * 16 + mrow) * 16 + ka);
            v2f b = *(const v2f*)(ks + (tn * 16 + mrow) * 16 + ka);
            acc = wmma4(a, b, acc);
        }
        for (int r = 0; r < 8; ++r)
            S[(tm * 16 + r + 8 * khi) * 258 + tn * 16 + mrow] = acc[r] * 0.25f;
    }
    __syncthreads();

    // row softmax (thread per row)
    {
        float* row = S + tid * 258;
        float mx = row[0];
        for (int j = 1; j < 256; ++j) mx = fmaxf(mx, row[j]);
        float s = 0.f;
        for (int j = 0; j < 256; ++j) { float e = __expf(row[j] - mx); row[j] = e; s += e; }
        float inv = 1.f / s;
        for (int j = 0; j < 256; ++j) row[j] *= inv;
    }
    __syncthreads();

    // O = S @ V  (16 M-tiles x 1 N-tile, K=256), loads pipelined one step
    for (int tile = wid; tile < 16; tile += 8) {
        const float* srow = S + (tile * 16 + mrow) * 258;
        v8f acc = {};
        int ka = 2 * khi;
        v2f a = *(const v2f*)(srow + ka);
        v2f b;
        b.x = vs[ka * 16 + mrow];
        b.y = vs[(ka + 1) * 16 + mrow];
        for (int k0 = 0; k0 < 252; k0 += 4) {
            int kn = k0 + 4 + 2 * khi;
            v2f a2 = *(const v2f*)(srow + kn);
            v2f b2;
            b2.x = vs[kn * 16 + mrow];
            b2.y = vs[(kn + 1) * 16 + mrow];
            acc = wmma4(a, b, acc);
            a = a2; b = b2;
        }
        acc = wmma4(a, b, acc);
        for (int r = 0; r < 8; ++r)
            obuf[(base + tile * 16 + r + 8 * khi) * 64 + h * 16 + mrow] = acc[r];
    }
}

// ===========================================================================
// Small glue kernels
// ===========================================================================
__global__ void fusion_kernel(const float* __restrict__ pf, const float* __restrict__ pb,
                              const float* __restrict__ jf, const float* __restrict__ jb,
                              const float* __restrict__ W1, const float* __restrict__ b1,
                              const float* __restrict__ W2, const float* __restrict__ b2,
                              float* __restrict__ speed0, float* __restrict__ jam0,
                              float* __restrict__ wgt, int nt)
{
    int i = blockIdx.x * blockDim.x + threadIdx.x;
    if (i >= nt) return;
    float a_pf = pf[i], a_pb = pb[i];
    float l0 = b2[0], l1 = b2[1];
    for (int h = 0; h < 64; ++h) {
        float a = fmaxf(W1[h * 2] * a_pf + W1[h * 2 + 1] * a_pb + b1[h], 0.f);
        l0 += W2[h] * a;
        l1 += W2[64 + h] * a;
    }
    float mx = fmaxf(l0, l1);
    float e0 = __expf(l0 - mx), e1 = __expf(l1 - mx);
    float inv = 1.f / (e0 + e1);
    float w0 = e0 * inv, w1 = e1 * inv;
    speed0[i] = w0 * a_pf + w1 * a_pb;
    jam0[i]   = w0 * jf[i] + w1 * jb[i];
    wgt[i * 2] = w0; wgt[i * 2 + 1] = w1;
}

__global__ void hfpos_kernel(const float* __restrict__ hsF, const float* __restrict__ hsB,
                             const float* __restrict__ wgt, float* __restrict__ x0, size_t total)
{
    size_t idx = (size_t)blockIdx.x * blockDim.x + threadIdx.x;
    if (idx >= total) return;
    int c = (int)(idx & 63);
    size_t r = idx >> 6;                  // r = n*256 + t
    int t = (int)(r & 255);
    float v = hsF[idx] * wgt[r * 2] + hsB[idx] * wgt[r * 2 + 1];
    int i2 = c & ~1;
    float div = __powf(10000.f, (float)i2 / 64.f);
    float ang = (float)t / div;
    v += (c & 1) ? __cosf(ang) : __sinf(ang);
    x0[idx] = v;
}

__global__ void ln_kernel(const float* __restrict__ X, const float* __restrict__ g,
                          const float* __restrict__ b, float* __restrict__ Y, int rows)
{
    int r = blockIdx.x * blockDim.x + threadIdx.x;
    if (r >= rows) return;
    const float* x = X + (size_t)r * 64;
    float mu = 0.f;
    for (int c = 0; c < 64; ++c) mu += x[c];
    mu *= (1.f / 64.f);
    float var = 0.f;
    for (int c = 0; c < 64; ++c) { float dd = x[c] - mu; var += dd * dd; }
    var *= (1.f / 64.f);
    float inv = rsqrtf(var + 1e-5f);
    float* y = Y + (size_t)r * 64;
    for (int c = 0; c < 64; ++c) y[c] = (x[c] - mu) * inv * g[c] + b[c];
}

__global__ void proj_kernel(const float* __restrict__ X, const float* __restrict__ pW,
                            const float* __restrict__ pb, float* __restrict__ out, int rows)
{
    int r = blockIdx.x * blockDim.x + threadIdx.x;
    if (r >= rows) return;
    const float* x = X + (size_t)r * 64;
    float s = pb[0];
    for (int c = 0; c < 64; ++c) s += x[c] * pW[c];
    out[r] = s;
}

__global__ void combine_kernel(const float* __restrict__ s0, const float* __restrict__ j0,
                               const float* __restrict__ sr, const float* __restrict__ jr,
                               float* __restrict__ out, int nt)
{
    int i = blockIdx.x * blockDim.x + threadIdx.x;
    if (i >= nt) return;
    out[i]      = 0.7f * s0[i] + 0.3f * sr[i];
    out[nt + i] = 0.7f * j0[i] + 0.3f * jr[i];
}

// ===========================================================================
// Host orchestration
// ===========================================================================
extern "C" void kernel_launch(void* const* d_in, const int* in_sizes, int n_in,
                              void* d_out, int out_size, void* d_ws, size_t ws_size,
                              hipStream_t stream)
{
    (void)in_sizes; (void)n_in; (void)out_size; (void)ws_size;

    const float* x_seq   = (const float*)d_in[0];
    const float* m_seq   = (const float*)d_in[1];
    const float* tf_seq  = (const float*)d_in[2];
    const float* tj_seq  = (const float*)d_in[3];
    const float* Anorm   = (const float*)d_in[4];
    const float* chebW0  = (const float*)d_in[5];
    const float* chebb0  = (const float*)d_in[6];
    const float* chebW1  = (const float*)d_in[7];
    const float* mixW    = (const float*)d_in[8];
    const float* mixb    = (const float*)d_in[9];
    const float* Wih     = (const float*)d_in[10];
    const float* Whh     = (const float*)d_in[11];
    const float* bih     = (const float*)d_in[12];
    const float* bhh     = (const float*)d_in[13];
    const float* outW    = (const float*)d_in[14];
    const float* outb    = (const float*)d_in[15];
    const float* jamW    = (const float*)d_in[16];
    const float* jamb    = (const float*)d_in[17];
    const float* fuse_W1 = (const float*)d_in[18];
    const float* fuse_b1 = (const float*)d_in[19];
    const float* fuse_W2 = (const float*)d_in[20];
    const float* fuse_b2 = (const float*)d_in[21];
    const float* qkvW    = (const float*)d_in[22];
    const float* qkvb_   = (const float*)d_in[23];
    const float* oW      = (const float*)d_in[24];
    const float* ob      = (const float*)d_in[25];
    const float* ln1g    = (const float*)d_in[26];
    const float* ln1b    = (const float*)d_in[27];
    const float* ff1W    = (const float*)d_in[28];
    const float* ff1b    = (const float*)d_in[29];
    const float* ff2W    = (const float*)d_in[30];
    const float* ff2b    = (const float*)d_in[31];
    const float* ln2g    = (const float*)d_in[32];
    const float* ln2b    = (const float*)d_in[33];
    const float* projW   = (const float*)d_in[34];
    const float* projb   = (const float*)d_in[35];

    const size_t NT = (size_t)NN * TT;      // 78592 tokens
    float* ws = (float*)d_ws;
    size_t off = 0;
    auto alloc = [&](size_t n) { float* p = ws + off; off += n; return p; };

    float* hsF    = alloc(NT * 64);
    float* hsB    = alloc(NT * 64);
    float* pf     = alloc(NT);
    float* pb     = alloc(NT);
    float* jf     = alloc(NT);
    float* jb     = alloc(NT);
    float* speed0 = alloc(NT);
    float* jam0   = alloc(NT);
    float* wgt    = alloc(NT * 2);
    float* sref   = alloc(NT);
    float* jref   = alloc(NT);
    float* Apad   = alloc((size_t)AP_R * AP_C);
    float* x0     = alloc(NT * 64);
    float* xA     = alloc(NT * 64);
    float* xB     = alloc(NT * 64);
    float* t1     = alloc(NT * 64);
    float* qkvbuf = alloc(NT * 192);
    float* obuf   = alloc(NT * 64);
    float* midbuf = alloc(NT * 256);

    const int cellLds = CELL_LDS_FLOATS * 4;
    const int attnLds = ATTN_LDS_FLOATS * 4;
    hipFuncSetAttribute((const void*)cell_scan_kernel,
                        hipFuncAttributeMaxDynamicSharedMemorySize, cellLds);
    hipFuncSetAttribute((const void*)attn_kernel,
                        hipFuncAttributeMaxDynamicSharedMemorySize, attnLds);

    // ---- Phase 0: pad Anorm to 320x308 ------------------------------------
    pad_anorm_kernel<<<(AP_R * AP_C + 255) / 256, 256, 0, stream>>>(Anorm, Apad);

    // ---- Phase 1: bidirectional persistent scan (both dirs in parallel) ---
    cell_scan_kernel<<<2, 1024, cellLds, stream>>>(
        x_seq, m_seq, tf_seq, tj_seq, Apad,
        chebW0, chebb0, chebW1, mixW, mixb,
        Wih, Whh, bih, bhh, outW, outb, jamW, jamb,
        hsF, hsB, pf, pb, jf, jb);

    // ---- Phase 2: fusion + positional encoding ----------------------------
    fusion_kernel<<<(int)((NT + 255) / 256), 256, 0, stream>>>(
        pf, pb, jf, jb, fuse_W1, fuse_b1, fuse_W2, fuse_b2,
        speed0, jam0, wgt, (int)NT);
    hfpos_kernel<<<(int)((NT * 64 + 255) / 256), 256, 0, stream>>>(
        hsF, hsB, wgt, x0, NT * 64);

    // ---- Phase 3: two transformer enhancers -------------------------------
    auto gemmGrid = [](int M, int Nc) { return ((M / 16) * (Nc / 64) + 7) / 8; };
    const int M = (int)NT;
    const int lnGrid = (int)((NT + 255) / 256);

    for (int e = 0; e < 2; ++e) {
        const float* cur = x0;
        for (int l = 0; l < 3; ++l) {
            const int wi = e * 3 + l;
            // qkv = cur @ qkvW^T + b
            gemm_xwt_kernel<<<gemmGrid(M, 192), 256, 0, stream>>>(
                cur, qkvW + (size_t)wi * 192 * 64, qkvb_ + (size_t)wi * 192,
                nullptr, qkvbuf, M, 64, 192, 0);
            // attention (per node, per head)
            attn_kernel<<<NN * 4, 256, attnLds, stream>>>(qkvbuf, obuf);
            // o-proj + residual
            gemm_xwt_kernel<<<gemmGrid(M, 64), 256, 0, stream>>>(
                obuf, oW + (size_t)wi * 64 * 64, ob + (size_t)wi * 64,
                cur, t1, M, 64, 64, 0);
            ln_kernel<<<lnGrid, 256, 0, stream>>>(
                t1, ln1g + (size_t)wi * 64, ln1b + (size_t)wi * 64, xA, M);
            // FF: mid = relu(xA @ f1W^T + b)
            gemm_xwt_kernel<<<gemmGrid(M, 256), 256, 0, stream>>>(
                xA, ff1W + (size_t)wi * 256 * 64, ff1b + (size_t)wi * 256,
                nullptr, midbuf, M, 64, 256, 1);
            // f = mid @ f2W^T + b + xA
            gemm_xwt_kernel<<<gemmGrid(M, 64), 256, 0, stream>>>(
                midbuf, ff2W + (size_t)wi * 64 * 256, ff2b + (size_t)wi * 64,
                xA, t1, M, 256, 64, 0);
            ln_kernel<<<lnGrid, 256, 0, stream>>>(
                t1, ln2g + (size_t)wi * 64, ln2b + (size_t)wi * 64, xB, M);
            cur = xB;
        }
        proj_kernel<<<lnGrid, 256, 0, stream>>>(
            xB, projW + (size_t)e * 64, projb + e, e ? jref : sref, M);
    }

    // ---- Phase 4: alpha blend to output -----------------------------------
    combine_kernel<<<lnGrid, 256, 0, stream>>>(
        speed0, jam0, sref, jref, (float*)d_out, (int)NT);
}